// RotateMyLayer_8469675508076
// MI455X (gfx1250) — compile-verified
//
#include <hip/hip_runtime.h>
#include <stdint.h>

// STN bilinear warp: out[10, 8, 384, 256] from feature_map[8,160,160,256].
// Memory-bound gather kernel for MI455X:
//  - 3D grid (no integer division), block-uniform ROI params via s_load
//  - theta[6] staged once per block via gfx1250 async-to-LDS
//    (global_load_async_to_lds_b32 + s_wait_asynccnt)
//  - b128 coalesced corner gathers (RT, L2-resident reuse)
//  - b128 non-temporal output stores (write-once stream, keep L2 for gathers)

#define OUT_H   8
#define MAX_W   384
#define MAX_ROI 10
#define FH      160
#define FW      160
#define FC      256
#define C4      (FC / 4)                 // 64 float4 per pixel
#define POS_PER_BLOCK 4
#define NTHREADS (POS_PER_BLOCK * C4)    // 256 threads = 8 waves (wave32)

typedef float v4f __attribute__((ext_vector_type(4)));

__global__ __launch_bounds__(NTHREADS) void stn_warp_kernel(
    const float* __restrict__ fm,      // [8,160,160,256]
    const float* __restrict__ thetas,  // [10,6]
    const int*   __restrict__ masks,   // [10]
    const int*   __restrict__ widths,  // [10]
    float* __restrict__ out)           // [10,8,384,256]
{
    __shared__ float sp[8];            // [0..5] = theta (8 for LDS bank alignment)

    const int tid    = threadIdx.x;
    const int xiBase = blockIdx.x * POS_PER_BLOCK;  // 0..380
    const int yi     = blockIdx.y;                  // 0..7
    const int roi    = blockIdx.z;                  // 0..9

    // Block-uniform ROI params -> scalar loads (s_load_b32)
    const int bmask = masks[roi];
    const int w     = widths[roi];

    // --- CDNA5 async load to LDS: stage 6 theta dwords for this ROI ---
    if (tid < 6) {
        uint64_t ga = (uint64_t)(uintptr_t)(thetas + roi * 6 + tid); // GV mode addr
        uint32_t la = (uint32_t)(uintptr_t)(&sp[tid]);               // LDS byte offset
        asm volatile("global_load_async_to_lds_b32 %0, %1, off"
                     :: "v"(la), "v"(ga) : "memory");
        asm volatile("s_wait_asynccnt 0" ::: "memory");              // ASYNCcnt == 0
    }
    __syncthreads();

    const float t00 = sp[0], t01 = sp[1], t02 = sp[2];
    const float t10 = sp[3], t11 = sp[4], t12 = sp[5];

    const int p  = tid >> 6;        // position within block (0..3)
    const int c4 = tid & (C4 - 1);  // channel quad (0..63)
    const int xi = xiBase + p;

    v4f res = (v4f)(0.0f);

    if (xi < w) {
        // sampling grid (STN convention)
        const float yt    = -1.0f + (2.0f / (float)(OUT_H - 1)) * (float)yi;
        const float denom = (float)((w - 1) > 1 ? (w - 1) : 1);
        const float xt    = -1.0f + 2.0f * (float)xi / denom;
        const float xs    = fmaf(t00, xt, fmaf(t01, yt, t02));
        const float ys    = fmaf(t10, xt, fmaf(t11, yt, t12));
        const float x     = (xs + 1.0f) * (0.5f * (float)FW);
        const float y     = (ys + 1.0f) * (0.5f * (float)FH);

        const float x0f = floorf(x), y0f = floorf(y);
        int x0 = (int)x0f, y0i = (int)y0f;
        int x1 = x0 + 1, y1 = y0i + 1;
        x0  = min(max(x0,  0), FW - 1);
        x1  = min(max(x1,  0), FW - 1);
        y0i = min(max(y0i, 0), FH - 1);
        y1  = min(max(y1,  0), FH - 1);

        const float lx = x - x0f, ly = y - y0f;
        const float wa = (1.0f - lx) * (1.0f - ly);
        const float wb = (1.0f - lx) * ly;
        const float wc = lx * (1.0f - ly);
        const float wd = lx * ly;

        // 4 coalesced b128 corner gathers: each corner is a contiguous 1KB
        // channel run covered by 64 consecutive lanes. Stays RT for L2 reuse.
        const v4f* __restrict__ fm4 = (const v4f*)fm;
        const long base = (long)bmask * (long)(FH * FW * C4);
        const v4f Ia = fm4[base + (long)(y0i * FW + x0) * C4 + c4];
        const v4f Ib = fm4[base + (long)(y1  * FW + x0) * C4 + c4];
        const v4f Ic = fm4[base + (long)(y0i * FW + x1) * C4 + c4];
        const v4f Id = fm4[base + (long)(y1  * FW + x1) * C4 + c4];

        res.x = fmaf(wa, Ia.x, fmaf(wb, Ib.x, fmaf(wc, Ic.x, wd * Id.x)));
        res.y = fmaf(wa, Ia.y, fmaf(wb, Ib.y, fmaf(wc, Ic.y, wd * Id.y)));
        res.z = fmaf(wa, Ia.z, fmaf(wb, Ib.z, fmaf(wc, Ic.z, wd * Id.z)));
        res.w = fmaf(wa, Ia.w, fmaf(wb, Ib.w, fmaf(wc, Ic.w, wd * Id.w)));
    }

    // Coalesced non-temporal b128 store (also zero-fills masked columns).
    const long pos = ((long)roi * OUT_H + yi) * MAX_W + xi;
    __builtin_nontemporal_store(res, (v4f*)out + pos * C4 + c4);
}

extern "C" void kernel_launch(void* const* d_in, const int* in_sizes, int n_in,
                              void* d_out, int out_size, void* d_ws, size_t ws_size,
                              hipStream_t stream) {
    (void)in_sizes; (void)n_in; (void)d_ws; (void)ws_size; (void)out_size;
    const float* fm     = (const float*)d_in[0];   // [8,160,160,256]
    const float* thetas = (const float*)d_in[1];   // [10,6]
    const int*   masks  = (const int*)d_in[2];     // [10]
    const int*   widths = (const int*)d_in[3];     // [10]
    float*       out    = (float*)d_out;           // [10,8,384,256]

    dim3 grid(MAX_W / POS_PER_BLOCK, OUT_H, MAX_ROI);   // (96, 8, 10)
    stn_warp_kernel<<<grid, NTHREADS, 0, stream>>>(fm, thetas, masks, widths, out);
}